// PixelContrastLoss_2430951490101
// MI455X (gfx1250) — compile-verified
//
#include <hip/hip_runtime.h>
#include <math.h>

typedef __attribute__((ext_vector_type(2))) float v2f;
typedef __attribute__((ext_vector_type(8))) float v8f;

constexpr int   kC        = 256;        // feature dim (reference: C=256)
constexpr int   kHW       = 128 * 128;  // pixels per image (reference: H=W=128)
constexpr float kTemp     = 0.1f;
constexpr float kBaseTemp = 0.07f;
constexpr int   kRowsWG   = 64;         // rows per workgroup (2 row-waves x 32 rows)
constexpr int   kColSeg   = 8;          // column segments (grid.y)

__device__ __forceinline__ void atomicMaxF32(float* addr, float v) {
  // float ordering == int ordering for v>=0, reversed-unsigned for v<0
  if (v >= 0.0f) atomicMax(reinterpret_cast<int*>(addr), __float_as_int(v));
  else           atomicMin(reinterpret_cast<unsigned int*>(addr), __float_as_uint(v));
}

__global__ void init_stats(float* rowmax, float* negsum, float* possum, int N) {
  int i = blockIdx.x * blockDim.x + threadIdx.x;
  if (i < N) {
    rowmax[i] = -__builtin_inff();
    negsum[i] = 0.0f;
    possum[i] = 0.0f;
  }
}

// cf[n*kC + c] = feats[img_idx[t]*kC*kHW + c*kHW + pix_idx[t*V + v]],  n = v*T + t
__global__ void gather_rows(const float* __restrict__ feats,
                            const int* __restrict__ img_idx,
                            const int* __restrict__ pix_idx,
                            float* __restrict__ cf, int T, int V) {
  const int n = blockIdx.x;
  const int t = n % T;
  const int v = n / T;
  const long base = (long)img_idx[t] * kC * kHW + (long)pix_idx[t * V + v];
  for (int c = threadIdx.x; c < kC; c += blockDim.x)
    cf[(long)n * kC + c] = feats[base + (long)c * kHW];
}

// One streaming pass over the Gram matrix G = (cf . cf^T) / kTemp.
// MODE 0: out = per-row max of G                       -> rowmax
// MODE 1: out += exp(g - rowmax) over negatives        -> negsum
// MODE 2: out += v - log(exp(v)+neg) over positives\diag, v = g - rowmax -> possum
//
// WG = 4 waves in a 2x2 arrangement: wy selects a 32-row half of the 64-row
// LDS A block, wx selects a 32-col half of a 64-col super-tile.  Each wave
// computes a 32x32 output tile per iteration: 2 A-frags + 2 B-frags feed
// 4 WMMAs per k-step (1 wmma : 1 load).
template <int MODE>
__global__ void gram_pass(const float* __restrict__ cf,
                          const int* __restrict__ labels_s,
                          const float* rowmax,
                          const float* negsum,
                          float* outv,
                          int N, int T) {
  __shared__ float ldsA[kRowsWG * kC];  // 64 KB, xor-swizzled
  const int tid  = threadIdx.x;
  const int lane = tid & 31;
  const int wave = tid >> 5;
  const int wy   = wave >> 1;           // which 32-row block
  const int wx   = wave & 1;            // which 32-col half of the super-tile
  const int half = lane >> 4;
  const int l15  = lane & 15;
  const int wgRow = blockIdx.x * kRowsWG;
  const int supersPerSeg = (N / 64) / kColSeg;
  const int super0 = blockIdx.y * supersPerSeg;
  const float invT = 1.0f / kTemp;

  // Stage A rows [wgRow, wgRow+64) into LDS.
  // Element (r,c) stored at r*kC + (c ^ ((r&15)<<2)) -> conflict-free frag reads.
  {
    const float4* src = reinterpret_cast<const float4*>(cf + (long)wgRow * kC);
    for (int i = tid; i < kRowsWG * (kC / 4); i += blockDim.x) {
      int r  = i / (kC / 4);
      int c4 = i % (kC / 4);
      float4 d = src[(long)r * (kC / 4) + c4];
      int c = (c4 * 4) ^ ((r & 15) << 2);            // xor touches bits 2..5 only
      *reinterpret_cast<float4*>(&ldsA[r * kC + c]) = d;
    }
  }
  __syncthreads();

  // C/D layout: VGPR r, lane L -> (row = r + 8*(L>=16), col = L&15).
  // Per-lane row metadata for the wave's 32 rows (2 sub-tiles x 8 each).
  const int waveRow = wgRow + wy * 32;
  int labr[2][8]; float rmr[2][8]; float ngr[2][8];
#pragma unroll
  for (int iy = 0; iy < 2; ++iy)
#pragma unroll
    for (int r = 0; r < 8; ++r) {
      const int rg = waveRow + iy * 16 + half * 8 + r;
      labr[iy][r] = labels_s[rg % T];
      rmr[iy][r] = 0.0f; ngr[iy][r] = 0.0f;
      if (MODE >= 1) rmr[iy][r] = rowmax[rg];
      if (MODE == 2) ngr[iy][r] = negsum[rg];
    }

  float acc8[2][8];                    // per-row stats (both col sub-tiles merge)
#pragma unroll
  for (int iy = 0; iy < 2; ++iy)
#pragma unroll
    for (int r = 0; r < 8; ++r) acc8[iy][r] = (MODE == 0) ? -__builtin_inff() : 0.0f;

  // A fragment: lane L reads row (L&15) of the 16-row sub-block,
  // K = k0 + 2*half + {0,1}  (ISA 32-bit 16x4 A layout).
  const int    aswz  = l15 << 2;
  const float* aRow0 = &ldsA[(wy * 32 + l15) * kC];
  const float* aRow1 = aRow0 + 16 * kC;

  for (int ct = 0; ct < supersPerSeg; ++ct) {
    const int colBase = (super0 + ct) * 64 + wx * 32;
    const int col0 = colBase + l15;
    const int col1 = col0 + 16;
    const int labc0 = labels_s[col0 % T];
    const int labc1 = labels_s[col1 % T];
    // B fragment (4x16): lane L supplies column N=L&15, K = k0 + 2*half + {0,1};
    // B[k][n] = cf[col][k] -> one cf row per lane, float2 per k-step.
    const float* bRow0 = cf + (long)col0 * kC + 2 * half;
    const float* bRow1 = cf + (long)col1 * kC + 2 * half;
    if (ct + 1 < supersPerSeg) {                      // hide L2 latency on B
      __builtin_prefetch(bRow0 + 64 * kC, 0, 0);
      __builtin_prefetch(bRow1 + 64 * kC, 0, 0);
    }

    v8f a00 = {0.f,0.f,0.f,0.f,0.f,0.f,0.f,0.f};
    v8f a01 = a00, a10 = a00, a11 = a00;
#pragma unroll 4
    for (int k0 = 0; k0 < kC; k0 += 4) {
      const int ka = (k0 + 2 * half) ^ aswz;
      v2f fa0; fa0.x = aRow0[ka]; fa0.y = aRow0[ka + 1];
      v2f fa1; fa1.x = aRow1[ka]; fa1.y = aRow1[ka + 1];
      v2f fb0; fb0.x = bRow0[k0]; fb0.y = bRow0[k0 + 1];
      v2f fb1; fb1.x = bRow1[k0]; fb1.y = bRow1[k0 + 1];
      a00 = __builtin_amdgcn_wmma_f32_16x16x4_f32(false, fa0, false, fb0, (short)0, a00, false, false);
      a01 = __builtin_amdgcn_wmma_f32_16x16x4_f32(false, fa0, false, fb1, (short)0, a01, false, false);
      a10 = __builtin_amdgcn_wmma_f32_16x16x4_f32(false, fa1, false, fb0, (short)0, a10, false, false);
      a11 = __builtin_amdgcn_wmma_f32_16x16x4_f32(false, fa1, false, fb1, (short)0, a11, false, false);
    }

    auto process = [&](const v8f& acc, int iy, int colg, int labc) {
#pragma unroll
      for (int r = 0; r < 8; ++r) {
        const float g = acc[r] * invT;
        if (MODE == 0) {
          acc8[iy][r] = fmaxf(acc8[iy][r], g);
        } else if (MODE == 1) {
          if (labc != labr[iy][r]) acc8[iy][r] += __expf(g - rmr[iy][r]);
        } else {
          const int rowg = waveRow + iy * 16 + half * 8 + r;
          if (labc == labr[iy][r] && colg != rowg) {
            const float v = g - rmr[iy][r];
            acc8[iy][r] += v - __logf(__expf(v) + ngr[iy][r]);
          }
        }
      }
    };
    process(a00, 0, col0, labc0);
    process(a01, 0, col1, labc1);
    process(a10, 1, col0, labc0);
    process(a11, 1, col1, labc1);
  }

  // Reduce over the 16 lanes of each half (these lanes hold N=0..15 of one row),
  // then one atomic per (row, column-segment, wx half), i.e. two per row per segment.
#pragma unroll
  for (int iy = 0; iy < 2; ++iy)
#pragma unroll
    for (int r = 0; r < 8; ++r) {
      float v = acc8[iy][r];
#pragma unroll
      for (int off = 1; off < 16; off <<= 1) {
        const float o = __shfl_xor(v, off, 32);
        v = (MODE == 0) ? fmaxf(v, o) : (v + o);
      }
      if (l15 == 0) {
        const int rowg = waveRow + iy * 16 + half * 8 + r;
        if (MODE == 0) atomicMaxF32(&outv[rowg], v);
        else           atomicAdd(&outv[rowg], v);
      }
    }
}

__global__ void finalize(const float* __restrict__ possum,
                         const int* __restrict__ labels_s,
                         float* __restrict__ out, int N, int T, int V) {
  __shared__ int   sl[512];
  __shared__ float rnum[256], rden[256];
  for (int t = threadIdx.x; t < T; t += blockDim.x) sl[t] = labels_s[t];
  __syncthreads();
  float num = 0.f, den = 0.f;
  for (int i = threadIdx.x; i < N; i += blockDim.x) {
    const int lab = sl[i % T];
    int cc = 0;
    for (int t = 0; t < T; ++t) cc += (sl[t] == lab) ? 1 : 0;
    const float cnt  = (float)(V * cc - 1);      // mask.sum(axis=1)
    const float mlpp = possum[i] / cnt;
    if (lab != 0) { num += mlpp; den += 1.f; }   // BG_ANCHOR == False
  }
  rnum[threadIdx.x] = num; rden[threadIdx.x] = den;
  __syncthreads();
  for (int s = blockDim.x / 2; s > 0; s >>= 1) {
    if ((int)threadIdx.x < s) {
      rnum[threadIdx.x] += rnum[threadIdx.x + s];
      rden[threadIdx.x] += rden[threadIdx.x + s];
    }
    __syncthreads();
  }
  if (threadIdx.x == 0) out[0] = -(kTemp / kBaseTemp) * (rnum[0] / rden[0]);
}

extern "C" void kernel_launch(void* const* d_in, const int* in_sizes, int n_in,
                              void* d_out, int out_size, void* d_ws, size_t ws_size,
                              hipStream_t stream) {
  const float* feats    = (const float*)d_in[0];
  const int*   img_idx  = (const int*)d_in[1];
  const int*   pix_idx  = (const int*)d_in[2];
  const int*   labels_s = (const int*)d_in[3];
  const int T = in_sizes[3];          // anchors
  const int N = in_sizes[2];          // T * V total rows
  const int V = N / T;

  float* cf     = (float*)d_ws;                 // N x kC fp32 (8 MB)
  float* rowmax = cf + (size_t)N * kC;
  float* negsum = rowmax + N;
  float* possum = negsum + N;
  float* out    = (float*)d_out;

  init_stats<<<(N + 255) / 256, 256, 0, stream>>>(rowmax, negsum, possum, N);
  gather_rows<<<N, 256, 0, stream>>>(feats, img_idx, pix_idx, cf, T, V);

  dim3 grid(N / kRowsWG, kColSeg);
  gram_pass<0><<<grid, 128, 0, stream>>>(cf, labels_s, rowmax, negsum, rowmax, N, T);
  gram_pass<1><<<grid, 128, 0, stream>>>(cf, labels_s, rowmax, negsum, negsum, N, T);
  gram_pass<2><<<grid, 128, 0, stream>>>(cf, labels_s, rowmax, negsum, possum, N, T);

  finalize<<<1, 256, 0, stream>>>(possum, labels_s, out, N, T, V);
}